// Simple_backbone_70188355551855
// MI455X (gfx1250) — compile-verified
//
#include <hip/hip_runtime.h>
#include <stdint.h>

// SimpleConv mean-aggregation, 3 layers, self-loops.
// out layout: [4, N, 128] f32 = [x, h1, h2, h3].
// Per layer l:
//   1) out[l+1] = out[l]                         (self term, TDM DMA copy)
//   2) out[l+1][dst] += out[l][src] per edge     (async-LDS gather + f32 atomics)
//   3) out[l+1][i]   *= inv_deg[i]
// inv_deg (N floats) lives in d_ws, rebuilt every call (ws is poisoned).

#define FEAT 128
#define WAVES_PER_BLOCK 8
#define EDGES_PER_WAVE 4
#define TDM_CHUNK 4096  // floats per TDM block (16 KB)

typedef uint32_t u32x4 __attribute__((ext_vector_type(4)));
typedef int      i32x4 __attribute__((ext_vector_type(4)));
typedef int      i32x8 __attribute__((ext_vector_type(8)));

// ------------------------------------------------------ async LDS helpers ---
__device__ __forceinline__ void async_g2lds_b128(uint32_t lds_byte_addr, const void* gaddr) {
    asm volatile("global_load_async_to_lds_b128 %0, %1, off"
                 :
                 : "v"(lds_byte_addr), "v"((uint64_t)(uintptr_t)gaddr)
                 : "memory");
}

__device__ __forceinline__ void wait_async0() {
    asm volatile("s_wait_asynccnt 0" ::: "memory");
}

__device__ __forceinline__ void wait_async_le1() {
    asm volatile("s_wait_asynccnt 1" ::: "memory");
}

__device__ __forceinline__ uint32_t lds_addr_of(const void* p) {
    // Generic LDS pointers carry the LDS byte offset in the low 32 bits.
    return (uint32_t)(uintptr_t)p;
}

// ---------------------------------------------------------------- degree ----
__global__ void zero_f32_kernel(float* __restrict__ p, long long n) {
    long long i = (long long)blockIdx.x * blockDim.x + threadIdx.x;
    if (i < n) p[i] = 0.0f;
}

__global__ void degree_kernel(const long long* __restrict__ dst, float* __restrict__ deg,
                              long long E) {
    long long e = (long long)blockIdx.x * blockDim.x + threadIdx.x;
    if (e >= E) return;
    __builtin_prefetch(dst + e + 8192, 0, 0);
    long long t = dst[e];
    unsafeAtomicAdd(&deg[t], 1.0f);
}

__global__ void invdeg_kernel(float* __restrict__ deg, long long n) {
    long long i = (long long)blockIdx.x * blockDim.x + threadIdx.x;
    if (i < n) deg[i] = 1.0f / (deg[i] + 1.0f);
}

// ------------------------------------------------------- TDM bulk copy ------
// One wave per block; each block DMA-copies TDM_CHUNK floats global->LDS->global
// via the Tensor Data Mover (tracked with TENSORcnt). D# is a 1-D tile:
// data_size=4B, tile_dim0=TDM_CHUNK, tensor dims sized to the chunk (no OOB).
__device__ __forceinline__ u32x4 tdm_group0(uint64_t gaddr, uint32_t lds_addr) {
    u32x4 g0;
    g0[0] = 1u;                                               // count=1, user desc
    g0[1] = lds_addr;                                         // lds_addr (bytes)
    g0[2] = (uint32_t)gaddr;                                  // global_addr[31:0]
    g0[3] = (uint32_t)((gaddr >> 32) & 0x01FFFFFFu) | (2u << 30);  // addr[56:32] | type=2
    return g0;
}

__device__ __forceinline__ i32x8 tdm_group1(void) {
    const uint32_t TD0 = TDM_CHUNK;
    i32x8 g1;
    g1[0] = (int)(2u << 16);                       // data_size=4B; mask/pad/iter=0
    g1[1] = (int)((TD0 & 0xFFFFu) << 16);          // tensor_dim0[15:0]  @ bits 63:48
    g1[2] = (int)((TD0 >> 16) | (1u << 16));       // tensor_dim0[31:16], tensor_dim1=1
    g1[3] = (int)(TD0 << 16);                      // tile_dim0 @ bits 127:112
    g1[4] = 0;                                     // tile_dim1=0, tile_dim2=0
    g1[5] = (int)TD0;                              // tensor_dim0_stride[31:0]
    g1[6] = 0;
    g1[7] = 0;
    return g1;
}

#if __has_include(<hip/amd_detail/amd_gfx1250_TDM.h>)
// amdgpu-toolchain (clang-23): 6-arg builtin
#define TDM_LOAD(g0, g1, g2, g3)  __builtin_amdgcn_tensor_load_to_lds((g0), (g1), (g2), (g3), (i32x8){0,0,0,0,0,0,0,0}, 0)
#define TDM_STORE(g0, g1, g2, g3) __builtin_amdgcn_tensor_store_from_lds((g0), (g1), (g2), (g3), (i32x8){0,0,0,0,0,0,0,0}, 0)
#else
// ROCm 7.2 (clang-22): 5-arg builtin
#define TDM_LOAD(g0, g1, g2, g3)  __builtin_amdgcn_tensor_load_to_lds((g0), (g1), (g2), (g3), 0)
#define TDM_STORE(g0, g1, g2, g3) __builtin_amdgcn_tensor_store_from_lds((g0), (g1), (g2), (g3), 0)
#endif

__global__ __launch_bounds__(32) void copy_tdm_kernel(const float* __restrict__ src,
                                                      float* __restrict__ dst) {
    __shared__ float buf[TDM_CHUNK];
    const uint64_t off = (uint64_t)blockIdx.x * (TDM_CHUNK * sizeof(float));
    const uint64_t sga = (uint64_t)(uintptr_t)src + off;
    const uint64_t dga = (uint64_t)(uintptr_t)dst + off;
    const uint32_t lds = lds_addr_of(buf);

    const u32x4 g0l = tdm_group0(sga, lds);
    const u32x4 g0s = tdm_group0(dga, lds);
    const i32x8 g1  = tdm_group1();
    const i32x4 gz  = {0, 0, 0, 0};

    TDM_LOAD(g0l, g1, gz, gz);                 // tensor_load_to_lds
    __builtin_amdgcn_s_wait_tensorcnt(0);
    TDM_STORE(g0s, g1, gz, gz);                // tensor_store_from_lds
    __builtin_amdgcn_s_wait_tensorcnt(0);
}

// --------------------------------------------------------------- scatter ----
// One wave per EDGES_PER_WAVE edges. Two LDS slots per wave: while edge k's
// 512B row is being accumulated with f32 atomics, edge k+1's row gather is
// already in flight (async loads complete in order; s_wait_asynccnt 1 waits
// for the older gather only).
__global__ void scatter_kernel(const float* __restrict__ h, float* __restrict__ hn,
                               const long long* __restrict__ ei, long long E) {
    __shared__ float lds[WAVES_PER_BLOCK * 2 * FEAT];  // 8 KB
    const int lane = threadIdx.x & 31;
    const int wv   = threadIdx.x >> 5;
    const long long base = ((long long)blockIdx.x * WAVES_PER_BLOCK + wv) * EDGES_PER_WAVE;
    if (base >= E) return;

    float* slot[2] = { &lds[(wv * 2 + 0) * FEAT], &lds[(wv * 2 + 1) * FEAT] };
    const uint32_t la[2] = { lds_addr_of(slot[0] + lane * 4), lds_addr_of(slot[1] + lane * 4) };

    __builtin_prefetch(ei + base + 16384, 0, 0);        // edge src stream ahead
    __builtin_prefetch(ei + E + base + 16384, 0, 0);    // edge dst stream ahead

    // prologue: gather row for edge 0 into slot 0
    {
        long long s0 = ei[base];
        async_g2lds_b128(la[0], h + s0 * FEAT + lane * 4);
    }

    for (int k = 0; k < EDGES_PER_WAVE; ++k) {
        const long long e = base + k;
        if (e >= E) break;
        const bool more = (k + 1 < EDGES_PER_WAVE) && (e + 1 < E);
        if (more) {
            long long sn = ei[e + 1];
            async_g2lds_b128(la[(k + 1) & 1], h + sn * FEAT + lane * 4);
            wait_async_le1();   // wait for edge k's gather; k+1 stays in flight
        } else {
            wait_async0();
        }
        const long long t = ei[E + e];
        const float4 v = *(const float4*)(slot[k & 1] + lane * 4);
        float* drow = hn + t * FEAT + (long long)lane * 4;
        unsafeAtomicAdd(drow + 0, v.x);
        unsafeAtomicAdd(drow + 1, v.y);
        unsafeAtomicAdd(drow + 2, v.z);
        unsafeAtomicAdd(drow + 3, v.w);
    }
}

// ----------------------------------------------------------------- scale ----
__global__ void scale_kernel(float* __restrict__ hn, const float* __restrict__ invdeg,
                             long long n4) {
    long long i = (long long)blockIdx.x * blockDim.x + threadIdx.x;
    if (i >= n4) return;
    long long node = i >> 5;  // 128/4 = 32 float4 per node
    float s = invdeg[node];
    float4 v = ((const float4*)hn)[i];
    v.x *= s; v.y *= s; v.z *= s; v.w *= s;
    ((float4*)hn)[i] = v;
}

// ------------------------------------------------------------------ host ----
extern "C" void kernel_launch(void* const* d_in, const int* in_sizes, int n_in,
                              void* d_out, int out_size, void* d_ws, size_t ws_size,
                              hipStream_t stream) {
    const float*     x  = (const float*)d_in[0];
    const long long* ei = (const long long*)d_in[1];

    const long long N  = (long long)in_sizes[0] / FEAT;
    const long long E  = (long long)in_sizes[1] / 2;
    const long long ND = N * FEAT;
    const long long n4 = ND / 4;

    float* out    = (float*)d_out;
    float* invdeg = (float*)d_ws;  // N floats

    const int TB = 256;
    dim3 blkN((unsigned)((N + TB - 1) / TB));
    dim3 blkE((unsigned)((E + TB - 1) / TB));
    dim3 blk4((unsigned)((n4 + TB - 1) / TB));
    const long long edges_per_blk = (long long)WAVES_PER_BLOCK * EDGES_PER_WAVE;
    dim3 blkS((unsigned)((E + edges_per_blk - 1) / edges_per_blk));
    dim3 blkT((unsigned)((ND + TDM_CHUNK - 1) / TDM_CHUNK));

    // degree -> inv_deg (workspace is poisoned: rebuild every call)
    zero_f32_kernel<<<blkN, TB, 0, stream>>>(invdeg, N);
    degree_kernel<<<blkE, TB, 0, stream>>>(ei + E, invdeg, E);
    invdeg_kernel<<<blkN, TB, 0, stream>>>(invdeg, N);

    // out[0] = x  (TDM DMA)
    copy_tdm_kernel<<<blkT, 32, 0, stream>>>(x, out);

    for (int l = 0; l < 3; ++l) {
        const float* h  = out + (long long)l * ND;
        float*       hn = out + (long long)(l + 1) * ND;
        copy_tdm_kernel<<<blkT, 32, 0, stream>>>(h, hn);        // self term
        scatter_kernel<<<blkS, TB, 0, stream>>>(h, hn, ei, E);  // neighbors
        scale_kernel<<<blk4, TB, 0, stream>>>(hn, invdeg, n4);  // * inv_deg
    }
}